// AstroSurveyGNN_68633577390196
// MI455X (gfx1250) — compile-verified
//
#include <hip/hip_runtime.h>
#include <hip/hip_bf16.h>

// ---------------------------------------------------------------------------
// AstroSurveyGNN for MI455X (gfx1250, wave32)
//   - KNN: wave-per-node scan over LDS-staged coord tiles, register top-8,
//     wave-merge via shfl_xor min-reduction + ballot.
//   - GEMMs: V_WMMA_F32_16X16X4_F32 (full fp32 tensor path), wave-per-16x16 tile.
//   - GCN aggregate: global_atomic_add_f32 scatter (everything L2-resident).
// ---------------------------------------------------------------------------

#define Npts  16000
#define FIN   16
#define HID   256
#define OUTD  128
#define KNN   8
#define NLAY  3
#define NEDGE (Npts * KNN)

typedef __attribute__((ext_vector_type(2))) float v2f;
typedef __attribute__((ext_vector_type(8))) float v8f;

// ---------------------------------------------------------------------------
// 1) KNN: one wave per query node, 8 waves (8 nodes) per 256-thread block.
//    All waves in the block share LDS coordinate tiles.
// ---------------------------------------------------------------------------
#define TILE 2048

__global__ void knn_kernel(const float* __restrict__ coords,
                           int* __restrict__ nbr) {
    __shared__ float sx[TILE], sy[TILE], sz[TILE];

    const int lane = threadIdx.x & 31;
    const int wave = threadIdx.x >> 5;
    const int i    = blockIdx.x * 8 + wave;   // query node (always < Npts)

    const float cx = coords[i * 3 + 0];
    const float cy = coords[i * 3 + 1];
    const float cz = coords[i * 3 + 2];

    // per-lane sorted top-8 (ascending distance)
    float bd[KNN];
    int   bi[KNN];
#pragma unroll
    for (int s = 0; s < KNN; ++s) { bd[s] = 3.4e38f; bi[s] = -1; }

    for (int t0 = 0; t0 < Npts; t0 += TILE) {
        const int cnt = (Npts - t0) < TILE ? (Npts - t0) : TILE;
        __syncthreads();
        for (int idx = threadIdx.x; idx < cnt; idx += 256) {
            sx[idx] = coords[(t0 + idx) * 3 + 0];
            sy[idx] = coords[(t0 + idx) * 3 + 1];
            sz[idx] = coords[(t0 + idx) * 3 + 2];
        }
        __syncthreads();

        for (int jj = lane; jj < cnt; jj += 32) {
            const int j = t0 + jj;
            const float dx = sx[jj] - cx;
            const float dy = sy[jj] - cy;
            const float dz = sz[jj] - cz;
            const float d2 = dx * dx + dy * dy + dz * dz;
            if (j != i && d2 < bd[KNN - 1]) {
                bd[KNN - 1] = d2; bi[KNN - 1] = j;
#pragma unroll
                for (int s = KNN - 1; s > 0; --s) {
                    if (bd[s] < bd[s - 1]) {
                        float td = bd[s]; bd[s] = bd[s - 1]; bd[s - 1] = td;
                        int   ti = bi[s]; bi[s] = bi[s - 1]; bi[s - 1] = ti;
                    }
                }
            }
        }
    }

    // wave-level merge of 32 sorted lists: 8 rounds of min-extract
#pragma unroll
    for (int r = 0; r < KNN; ++r) {
        const float front = bd[0];
        float m = front;
#pragma unroll
        for (int off = 16; off > 0; off >>= 1)
            m = fminf(m, __shfl_xor(m, off, 32));
        const unsigned mask = (unsigned)__ballot(front == m);
        const int winner = __ffs((int)mask) - 1;
        const int jidx   = __shfl(bi[0], winner, 32);
        if (lane == 0) nbr[i * KNN + r] = jidx;
        if (lane == winner) {
#pragma unroll
            for (int s = 0; s < KNN - 1; ++s) { bd[s] = bd[s + 1]; bi[s] = bi[s + 1]; }
            bd[KNN - 1] = 3.4e38f; bi[KNN - 1] = -1;
        }
    }
}

// ---------------------------------------------------------------------------
// 2) degree / normalization
// ---------------------------------------------------------------------------
__global__ void deg_init_kernel(float* __restrict__ deg) {
    const int i = blockIdx.x * blockDim.x + threadIdx.x;
    if (i < Npts) deg[i] = 1.0f;
}

__global__ void deg_count_kernel(const int* __restrict__ nbr,
                                 float* __restrict__ deg) {
    const int e = blockIdx.x * blockDim.x + threadIdx.x;
    if (e < NEDGE) atomicAdd(&deg[nbr[e]], 1.0f);
}

__global__ void norm_kernel(float* __restrict__ deg) {
    const int i = blockIdx.x * blockDim.x + threadIdx.x;
    if (i < Npts) deg[i] = rsqrtf(deg[i]);   // in-place -> norm
}

// ---------------------------------------------------------------------------
// 3) fp32 WMMA GEMM:  C[M,N] = A[M,K] @ B[K,N] (+ bias[N])
//    One wave per 16x16 tile of C; V_WMMA_F32_16X16X4_F32, K stepped by 4.
//    A layout (16x4 f32): lane<16 -> row M=lane, K=k0,k1; lane>=16 -> K=k2,k3
//    B layout (4x16 f32): lane<16 -> col N=lane, K=k0,k1; lane>=16 -> K=k2,k3
//    C/D layout: VGPR r -> M = r + 8*(lane>>4), N = lane&15
// ---------------------------------------------------------------------------
__global__ void gemm_wmma_f32(const float* __restrict__ A,
                              const float* __restrict__ B,
                              const float* __restrict__ bias,
                              float* __restrict__ C,
                              int M, int Nn, int Kd) {
    const int lane = threadIdx.x & 31;
    const int wave = threadIdx.x >> 5;
    const int ntiles = Nn >> 4;
    const int mtiles = M >> 4;
    const int tile = blockIdx.x * (blockDim.x >> 5) + wave;
    if (tile >= ntiles * mtiles) return;     // uniform per wave: EXEC stays all-1s

    const int mt = tile / ntiles;
    const int nt = tile - mt * ntiles;
    const int m0 = mt << 4, n0 = nt << 4;
    const int half = lane >> 4;              // 0: K pair 0/1, 1: K pair 2/3
    const int l15  = lane & 15;

    const float* arow = A + (size_t)(m0 + l15) * Kd + half * 2;
    const float* bcol = B + (size_t)(half * 2) * Nn + n0 + l15;

    v8f c = {};
#pragma unroll 2
    for (int k = 0; k < Kd; k += 4) {
        v2f a = *(const v2f*)(arow + k);                 // K = k+2h, k+2h+1
        v2f b;
        b.x = bcol[(size_t)k * Nn];                      // row K = k+2h
        b.y = bcol[(size_t)(k + 1) * Nn];                // row K = k+2h+1
        c = __builtin_amdgcn_wmma_f32_16x16x4_f32(
                false, a, false, b, (short)0, c, false, false);
    }

    if (bias) {
        const float bv = bias[n0 + l15];
#pragma unroll
        for (int r = 0; r < 8; ++r) c[r] += bv;
    }

    float* crow = C + n0 + l15;
#pragma unroll
    for (int r = 0; r < 8; ++r)
        crow[(size_t)(m0 + r + 8 * half) * Nn] = c[r];
}

// ---------------------------------------------------------------------------
// 4) GCN aggregate: self term + bias, edge scatter (atomics), ReLU
// ---------------------------------------------------------------------------
__global__ void self_bias_kernel(const float* __restrict__ hw,
                                 const float* __restrict__ norm,
                                 const float* __restrict__ b,
                                 float* __restrict__ agg) {
    const int idx = blockIdx.x * blockDim.x + threadIdx.x;   // < Npts*HID
    const int i = idx >> 8;
    const int c = idx & 255;
    const float nv = norm[i];
    agg[idx] = hw[idx] * nv * nv + b[c];
}

__global__ void scatter_kernel(const float* __restrict__ hw,
                               const int* __restrict__ nbr,
                               const float* __restrict__ norm,
                               float* __restrict__ agg) {
    const int lane = threadIdx.x & 31;
    const int wave = threadIdx.x >> 5;
    const int e = blockIdx.x * 8 + wave;     // < NEDGE
    const int i = e >> 3;                    // src node
    const int j = nbr[e];                    // dst node
    const float s = norm[i] * norm[j];
    const float* hrow = hw  + (size_t)i * HID;
    float*       arow = agg + (size_t)j * HID;
#pragma unroll
    for (int r = 0; r < 8; ++r) {
        const int c = lane + 32 * r;
        atomicAdd(&arow[c], hrow[c] * s);
    }
}

__global__ void relu_kernel(const float* __restrict__ agg,
                            float* __restrict__ h) {
    const int idx = blockIdx.x * blockDim.x + threadIdx.x;   // < Npts*HID
    h[idx] = fmaxf(agg[idx], 0.0f);
}

// ---------------------------------------------------------------------------
// 5) mean pool + MLP head
// ---------------------------------------------------------------------------
__global__ void zero_kernel(float* __restrict__ g) {
    g[threadIdx.x] = 0.0f;                   // 256 threads
}

__global__ void pool_kernel(const float* __restrict__ h,
                            float* __restrict__ g_sum) {
    const int c  = threadIdx.x;              // 256 threads -> one column each
    const int r0 = blockIdx.x * 64;
    float s = 0.0f;
    for (int r = 0; r < 64; ++r) s += h[(size_t)(r0 + r) * HID + c];
    atomicAdd(&g_sum[c], s);
}

__global__ void head_kernel(const float* __restrict__ g_sum,
                            const float* __restrict__ Wp1,
                            const float* __restrict__ bp1,
                            const float* __restrict__ Wp2,
                            const float* __restrict__ bp2,
                            float* __restrict__ out) {
    __shared__ float gs[HID];
    __shared__ float p1[HID / 2];
    const int t = threadIdx.x;               // 128 threads
    const float invN = 1.0f / (float)Npts;
    gs[t]       = g_sum[t]       * invN;
    gs[t + 128] = g_sum[t + 128] * invN;
    __syncthreads();

    float acc = bp1[t];
    for (int c = 0; c < HID; ++c) acc += gs[c] * Wp1[c * (HID / 2) + t];
    p1[t] = fmaxf(acc, 0.0f);
    __syncthreads();

    float o = bp2[t];
    for (int j = 0; j < HID / 2; ++j) o += p1[j] * Wp2[j * OUTD + t];
    out[t] = o;
}

// ---------------------------------------------------------------------------
// launch
// ---------------------------------------------------------------------------
extern "C" void kernel_launch(void* const* d_in, const int* in_sizes, int n_in,
                              void* d_out, int out_size, void* d_ws, size_t ws_size,
                              hipStream_t stream) {
    const float* x      = (const float*)d_in[0];
    const float* coords = (const float*)d_in[1];
    const float* We     = (const float*)d_in[2];
    const float* be     = (const float*)d_in[3];
    const float* gnn_W  = (const float*)d_in[4];
    const float* gnn_b  = (const float*)d_in[5];
    const float* Wp1    = (const float*)d_in[6];
    const float* bp1    = (const float*)d_in[7];
    const float* Wp2    = (const float*)d_in[8];
    const float* bp2    = (const float*)d_in[9];
    float* out = (float*)d_out;

    // workspace carve-up (256B aligned)
    char* ws = (char*)d_ws;
    size_t off = 0;
    auto carve = [&](size_t bytes) -> char* {
        char* p = ws + off;
        off += (bytes + 255) & ~(size_t)255;
        return p;
    };
    int*   nbr   = (int*)  carve((size_t)NEDGE * sizeof(int));
    float* norm  = (float*)carve((size_t)Npts * sizeof(float));   // deg -> norm
    float* h     = (float*)carve((size_t)Npts * HID * sizeof(float));
    float* hw    = (float*)carve((size_t)Npts * HID * sizeof(float));
    float* agg   = (float*)carve((size_t)Npts * HID * sizeof(float));
    float* g_sum = (float*)carve((size_t)HID * sizeof(float));

    // 1) graph construction
    knn_kernel<<<Npts / 8, 256, 0, stream>>>(coords, nbr);
    deg_init_kernel <<<(Npts  + 255) / 256, 256, 0, stream>>>(norm);
    deg_count_kernel<<<(NEDGE + 255) / 256, 256, 0, stream>>>(nbr, norm);
    norm_kernel     <<<(Npts  + 255) / 256, 256, 0, stream>>>(norm);

    // 2) encoder: h = x @ We + be   (M=16000, N=256, K=16)
    {
        const int tiles = (Npts / 16) * (HID / 16);
        gemm_wmma_f32<<<(tiles + 3) / 4, 128, 0, stream>>>(x, We, be, h,
                                                           Npts, HID, FIN);
    }

    // 3) GCN layers
    const int elems = Npts * HID;
    for (int l = 0; l < NLAY; ++l) {
        const int tiles = (Npts / 16) * (HID / 16);
        gemm_wmma_f32<<<(tiles + 3) / 4, 128, 0, stream>>>(
            h, gnn_W + (size_t)l * HID * HID, nullptr, hw, Npts, HID, HID);
        self_bias_kernel<<<elems / 256, 256, 0, stream>>>(
            hw, norm, gnn_b + (size_t)l * HID, agg);
        scatter_kernel<<<NEDGE / 8, 256, 0, stream>>>(hw, nbr, norm, agg);
        relu_kernel<<<elems / 256, 256, 0, stream>>>(agg, h);
    }

    // 4) mean pool + head
    zero_kernel<<<1, 256, 0, stream>>>(g_sum);
    pool_kernel<<<Npts / 64, 256, 0, stream>>>(h, g_sum);
    head_kernel<<<1, 128, 0, stream>>>(g_sum, Wp1, bp1, Wp2, bp2, out);
}